// GCNGraphNet_IMDB_34832184770971
// MI455X (gfx1250) — compile-verified
//
#include <hip/hip_runtime.h>
#include <hip/hip_bf16.h>
#include <math.h>

typedef __attribute__((ext_vector_type(2))) float v2f;
typedef __attribute__((ext_vector_type(8))) float v8f;

#define TPB 256

// ---------------------------------------------------------------- zero fill
__global__ void k_zero(float* __restrict__ p, long n) {
  long i = (long)blockIdx.x * blockDim.x + threadIdx.x;
  if (i < n) p[i] = 0.0f;
}

// ------------------------------------------------- degree atomics (per edge)
// degF[v] = #(row==v) + #(col==v)   (feature "degree" as in the torch code)
// degC[v] = #(col==v)               (GCN normalization degree, +1 self-loop later)
__global__ void k_edge_deg(const int* __restrict__ row, const int* __restrict__ col,
                           float* __restrict__ degF, float* __restrict__ degC, int E) {
  int e = blockIdx.x * blockDim.x + threadIdx.x;
  if (e >= E) return;
  int r = row[e], c = col[e];
  atomicAdd(&degF[r], 1.0f);
  atomicAdd(&degF[c], 1.0f);
  atomicAdd(&degC[c], 1.0f);
}

// ------------------------------------------------------------ dinv per node
__global__ void k_dinv(const float* __restrict__ degC, float* __restrict__ dinv, int N) {
  int v = blockIdx.x * blockDim.x + threadIdx.x;
  if (v >= N) return;
  dinv[v] = rsqrtf(degC[v] + 1.0f);   // deg always >= 1 (self-loop)
}

// ------------------------------------------- layer 1: hs = dinv * (x0 @ W1)
// x0 = [1, degF, rand, 0-pad] per node (K=4), W1 is 3x16 (K row 3 zero-padded).
// One V_WMMA_F32_16X16X4_F32 per 16-node tile per wave.
__global__ void k_gemm1_wmma(const float* __restrict__ degF,
                             const float* __restrict__ rand_feat,
                             const float* __restrict__ W1,
                             const float* __restrict__ dinv,
                             float* __restrict__ hs, int N) {
  int lane  = threadIdx.x & 31;
  int wave  = threadIdx.x >> 5;
  int node0 = (blockIdx.x * (blockDim.x >> 5) + wave) * 16;
  if (node0 >= N) return;                 // wave-uniform: EXEC stays all-1s
  int half = lane >> 4;                   // 0: K=0,1   1: K=2,3
  int l    = lane & 15;
  int m    = node0 + l;

  // A 16x4 fragment (ISA layout): lanes0-15 = (K0,K1), lanes16-31 = (K2,K3)
  float ax = half ? rand_feat[m] : 1.0f;
  float ay = half ? 0.0f         : degF[m];
  // B 4x16 fragment: lanes0-15 = rows K0,K1 ; lanes16-31 = rows K2,K3
  float bx = half ? W1[2 * 16 + l] : W1[0 * 16 + l];
  float by = half ? 0.0f           : W1[1 * 16 + l];
  v2f a = {ax, ay};
  v2f b = {bx, by};
  v8f c = {};
  c = __builtin_amdgcn_wmma_f32_16x16x4_f32(false, a, false, b, (short)0, c, false, false);

  // D layout: VGPR r -> row r (lanes 0-15) / row r+8 (lanes 16-31), col = lane&15
#pragma unroll
  for (int r = 0; r < 8; ++r) {
    int rr = node0 + r + half * 8;
    hs[(long)rr * 16 + l] = dinv[rr] * c[r];
  }
}

// -------------------------------- layer 2: out = dinv * (X @ W)  (16x16x16)
// 4 chained V_WMMA_F32_16X16X4_F32 per 16-node tile.
__global__ void k_gemm2_wmma(const float* __restrict__ X,
                             const float* __restrict__ W,
                             const float* __restrict__ dinv,
                             float* __restrict__ out, int N) {
  int lane  = threadIdx.x & 31;
  int wave  = threadIdx.x >> 5;
  int node0 = (blockIdx.x * (blockDim.x >> 5) + wave) * 16;
  if (node0 >= N) return;                 // wave-uniform
  int half = lane >> 4;
  int l    = lane & 15;
  int m    = node0 + l;

  v8f c = {};
#pragma unroll
  for (int k0 = 0; k0 < 16; k0 += 4) {
    int ka = k0 + half * 2;
    v2f a = { X[(long)m * 16 + ka], X[(long)m * 16 + ka + 1] };
    v2f b = { W[ka * 16 + l],       W[(ka + 1) * 16 + l]     };
    c = __builtin_amdgcn_wmma_f32_16x16x4_f32(false, a, false, b, (short)0, c, false, false);
  }
#pragma unroll
  for (int r = 0; r < 8; ++r) {
    int rr = node0 + r + half * 8;
    out[(long)rr * 16 + l] = dinv[rr] * c[r];
  }
}

// ------------------------------------ edge scatter: agg[col] += hs[row]
// lane-per-(edge,feature): 16 consecutive lanes -> 16 consecutive atomics
__global__ void k_scatter(const int* __restrict__ row, const int* __restrict__ col,
                          const float* __restrict__ src, float* __restrict__ agg,
                          long ET) {
  long t = (long)blockIdx.x * blockDim.x + threadIdx.x;
  if (t >= ET) return;
  int e = (int)(t >> 4);
  int f = (int)(t & 15);
  int r = row[e];
  int c = col[e];
  atomicAdd(&agg[(long)c * 16 + f], src[(long)r * 16 + f]);
}

// ------------------------- finalize layer1: x1 = relu(dinv*(agg+hs) + b1)
// writes x1 in-place over hs, and resets agg to 0 for the next layer
__global__ void k_fin1(float* __restrict__ hs, float* __restrict__ agg,
                       const float* __restrict__ dinv, const float* __restrict__ bias,
                       long NT) {
  long t = (long)blockIdx.x * blockDim.x + threadIdx.x;
  if (t >= NT) return;
  int v = (int)(t >> 4);
  int f = (int)(t & 15);
  float val = dinv[v] * (agg[t] + hs[t]) + bias[f];
  hs[t]  = val > 0.0f ? val : 0.0f;
  agg[t] = 0.0f;
}

// ---------------- finalize layer2 + mean-pool accumulation (batch atomics)
__global__ void k_fin2_pool(const float* __restrict__ hs, const float* __restrict__ agg,
                            const float* __restrict__ dinv, const float* __restrict__ bias,
                            const int* __restrict__ batch,
                            float* __restrict__ pool, float* __restrict__ cnt, long NT) {
  long t = (long)blockIdx.x * blockDim.x + threadIdx.x;
  if (t >= NT) return;
  int v = (int)(t >> 4);
  int f = (int)(t & 15);
  float val = dinv[v] * (agg[t] + hs[t]) + bias[f];
  int g = batch[v];
  atomicAdd(&pool[(long)g * 16 + f], val);
  if (f == 0) atomicAdd(&cnt[g], 1.0f);
}

// ------------------------------------------- mean + log_softmax per graph
__global__ void k_logsoftmax(const float* __restrict__ pool, const float* __restrict__ cnt,
                             float* __restrict__ out, int G) {
  int g = blockIdx.x * blockDim.x + threadIdx.x;
  if (g >= G) return;
  float inv = 1.0f / fmaxf(cnt[g], 1.0f);
  float x[16];
  float mx = -INFINITY;
#pragma unroll
  for (int f = 0; f < 16; ++f) {
    x[f] = pool[(long)g * 16 + f] * inv;
    mx = fmaxf(mx, x[f]);
  }
  float s = 0.0f;
#pragma unroll
  for (int f = 0; f < 16; ++f) s += expf(x[f] - mx);
  float lse = logf(s);
#pragma unroll
  for (int f = 0; f < 16; ++f) out[(long)g * 16 + f] = x[f] - mx - lse;
}

// ---------------------------------------------------------------------------
extern "C" void kernel_launch(void* const* d_in, const int* in_sizes, int n_in,
                              void* d_out, int out_size, void* d_ws, size_t ws_size,
                              hipStream_t stream) {
  const float* rand_feat = (const float*)d_in[0];
  const float* W1        = (const float*)d_in[1];
  const float* b1        = (const float*)d_in[2];
  const float* W2        = (const float*)d_in[3];
  const float* b2        = (const float*)d_in[4];
  const int*   edge      = (const int*)d_in[5];   // [2, E] flattened
  const int*   batch     = (const int*)d_in[6];   // [N]
  const int N = in_sizes[6];
  const int E = in_sizes[5] / 2;
  const int G = out_size / 16;
  const int* row = edge;
  const int* col = edge + E;
  float* out = (float*)d_out;

  // workspace layout (floats)
  float* ws   = (float*)d_ws;
  float* degF = ws;                         // N
  float* degC = degF + N;                   // N
  float* pool = degC + N;                   // G*16
  float* cnt  = pool + (size_t)G * 16;      // G
  float* dinv = cnt + G;                    // N
  float* hsA  = dinv + N;                   // N*16
  float* hsB  = hsA + (size_t)N * 16;       // N*16
  float* agg  = hsB + (size_t)N * 16;       // N*16
  (void)ws_size; (void)n_in;

  const long NT = (long)N * 16;
  const long ET = (long)E * 16;
  const long Z1 = 2L * N + (long)G * 17;    // degF, degC, pool, cnt (contiguous)

  auto blocks = [](long n) { return (unsigned)((n + TPB - 1) / TPB); };

  // 1) zero all accumulators (every call -> deterministic under graph replay)
  k_zero<<<blocks(Z1), TPB, 0, stream>>>(degF, Z1);
  k_zero<<<blocks(NT), TPB, 0, stream>>>(agg, NT);

  // 2) degrees
  k_edge_deg<<<blocks(E), TPB, 0, stream>>>(row, col, degF, degC, E);
  k_dinv<<<blocks(N), TPB, 0, stream>>>(degC, dinv, N);

  // 3) layer 1 transform (WMMA f32 16x16x4), hsA = dinv * (x0 @ W1)
  const int wavesNeeded = (N + 15) / 16;            // one 16-node tile per wave
  const int wavesPerBlk = TPB / 32;
  const unsigned gemmBlocks = (unsigned)((wavesNeeded + wavesPerBlk - 1) / wavesPerBlk);
  k_gemm1_wmma<<<gemmBlocks, TPB, 0, stream>>>(degF, rand_feat, W1, dinv, hsA, N);

  // 4) edge scatter + finalize (ReLU), agg reset inside finalize
  k_scatter<<<blocks(ET), TPB, 0, stream>>>(row, col, hsA, agg, ET);
  k_fin1<<<blocks(NT), TPB, 0, stream>>>(hsA, agg, dinv, b1, NT);

  // 5) layer 2 transform (WMMA), hsB = dinv * (x1 @ W2)
  k_gemm2_wmma<<<gemmBlocks, TPB, 0, stream>>>(hsA, W2, dinv, hsB, N);

  // 6) edge scatter + finalize + pooled accumulation
  k_scatter<<<blocks(ET), TPB, 0, stream>>>(row, col, hsB, agg, ET);
  k_fin2_pool<<<blocks(NT), TPB, 0, stream>>>(hsB, agg, dinv, b2, batch, pool, cnt, NT);

  // 7) mean + log_softmax
  k_logsoftmax<<<blocks(G), TPB, 0, stream>>>(pool, cnt, out, G);
}